// SGC_45483703665400
// MI455X (gfx1250) — compile-verified
//
#include <hip/hip_runtime.h>

typedef float v2f __attribute__((ext_vector_type(2)));
typedef float v8f __attribute__((ext_vector_type(8)));

#define FEAT 64

// Hardware fp32 atomic add, device scope (RMW resolves in the 192MB L2).
// Fire-and-forget (no return): tracked by STOREcnt, drained by the implicit
// S_WAIT_IDLE at S_ENDPGM, so kernel-boundary ordering is preserved.
__device__ __forceinline__ void atomic_add_f32_dev(float* p, float v) {
    asm volatile("global_atomic_add_f32 %0, %1, off scope:SCOPE_DEV"
                 :: "v"(p), "v"(v)
                 : "memory");
}

// ---------------------------------------------------------------------------
// Degree / normalization kernels
// ---------------------------------------------------------------------------
__global__ void deg_init_kernel(float* __restrict__ deg, int n) {
    int i = blockIdx.x * blockDim.x + threadIdx.x;
    if (i < n) deg[i] = 1.0f;  // self loop contributes 1 to every degree
}

__global__ void deg_count_kernel(const long long* __restrict__ col,
                                 float* __restrict__ deg, int e) {
    int i = blockIdx.x * blockDim.x + threadIdx.x;
    if (i < e) atomic_add_f32_dev(&deg[(int)col[i]], 1.0f);
}

__global__ void dinv_kernel(float* __restrict__ deg, int n) {
    int i = blockIdx.x * blockDim.x + threadIdx.x;
    if (i < n) {
        float d = deg[i];          // always >= 1 (self loop), no zero guard needed
        deg[i] = rsqrtf(d);        // store deg^{-1/2} in place
    }
}

// ---------------------------------------------------------------------------
// Propagation: hnext[i] = dinv[i]^2 * hcur[i]  (self-loop term, also zero-init)
// ---------------------------------------------------------------------------
__global__ void prop_init_kernel(const float* __restrict__ hcur,
                                 const float* __restrict__ dinv,
                                 float* __restrict__ hnext, int n) {
    int idx = blockIdx.x * blockDim.x + threadIdx.x;  // over n*FEAT
    if (idx < n * FEAT) {
        int i = idx >> 6;
        float w = dinv[i];
        hnext[idx] = (w * w) * hcur[idx];
    }
}

// hnext[row] += dinv[row]*dinv[col] * hcur[col]
// 64 threads per edge: coalesced 64-float gather of hcur[col] (4 cachelines),
// 64 hardware fp32 atomic adds into hnext[row] at L2.
// Each 32-lane wave lies wholly inside one 64-thread edge group, so the edge
// index is wave-uniform: readfirstlane forces scalarization of the index
// loads, dinv gathers, and the atomic base address.
__global__ void prop_edge_kernel(const long long* __restrict__ row,
                                 const long long* __restrict__ col,
                                 const float* __restrict__ dinv,
                                 const float* __restrict__ hcur,
                                 float* __restrict__ hnext, int e) {
    int tid = threadIdx.x;
    int f = tid & (FEAT - 1);
    long long eidx = (long long)blockIdx.x * (blockDim.x / FEAT) + (tid >> 6);
    if (eidx < e) {                       // wave-uniform branch
        int r = __builtin_amdgcn_readfirstlane((int)row[eidx]);
        int c = __builtin_amdgcn_readfirstlane((int)col[eidx]);
        float w = dinv[r] * dinv[c];      // scalar loads (uniform address)
        float msg = w * hcur[c * FEAT + f];
        atomic_add_f32_dev(&hnext[r * FEAT + f], msg);
    }
}

// ---------------------------------------------------------------------------
// out = h @ W^T + b  via V_WMMA_F32_16X16X4_F32 (full fp32 precision).
// One wave per 16-row strip, 4 accumulator tiles cover all N=64 output cols.
// Safe to run in-place (out == h): each wave finishes all loads of its 16
// rows (consumed by the last WMMA) before its stores.
// A 16x4 layout (ISA 7.12.2): lanes 0-15 -> M=lane, K={k0,k0+1};
//                             lanes 16-31 -> M=lane-16, K={k0+2,k0+3}.
// B 4x16 assumed mirror layout: v0 = B[k0+2*half][n], v1 = B[k0+2*half+1][n].
// D 16x16: VGPR r -> M = r (+8 for lanes 16-31), N = lane & 15.
// ---------------------------------------------------------------------------
#define WLDS_STRIDE 68  // pad: bank = (4*n + k) % 64, conflict-free, 8B aligned

__global__ void __launch_bounds__(256)
sgc_gemm_wmma_kernel(const float* __restrict__ h, const float* __restrict__ W,
                     const float* __restrict__ bias, float* __restrict__ out,
                     int n) {
    __shared__ float Wl[FEAT * WLDS_STRIDE];
    __shared__ float bl[FEAT];

    int tid = threadIdx.x;
    for (int i = tid; i < FEAT * FEAT; i += 256) {
        Wl[(i >> 6) * WLDS_STRIDE + (i & 63)] = W[i];  // W[j][k], j = out col
    }
    if (tid < FEAT) bl[tid] = bias[tid];
    __syncthreads();

    int wave = tid >> 5;
    int lane = tid & 31;
    int l    = lane & 15;
    int kh   = (lane >> 4) << 1;            // 0 for lanes 0-15, 2 for 16-31
    int m0   = blockIdx.x * 128 + wave * 16;
    if (m0 + 16 > n) return;                // wave-uniform: EXEC stays full

    const float* hrow = h + (long long)(m0 + l) * FEAT;
    const float* w0 = &Wl[( 0 + l) * WLDS_STRIDE];
    const float* w1 = &Wl[(16 + l) * WLDS_STRIDE];
    const float* w2 = &Wl[(32 + l) * WLDS_STRIDE];
    const float* w3 = &Wl[(48 + l) * WLDS_STRIDE];

    v8f acc0 = {}, acc1 = {}, acc2 = {}, acc3 = {};

#pragma unroll
    for (int k0 = 0; k0 < FEAT; k0 += 4) {
        v2f a  = *(const v2f*)(hrow + k0 + kh);   // A fragment (global)
        v2f b0 = *(const v2f*)(w0 + k0 + kh);     // B fragments (LDS, b64)
        v2f b1 = *(const v2f*)(w1 + k0 + kh);
        v2f b2 = *(const v2f*)(w2 + k0 + kh);
        v2f b3 = *(const v2f*)(w3 + k0 + kh);
        acc0 = __builtin_amdgcn_wmma_f32_16x16x4_f32(false, a, false, b0,
                                                     (short)0, acc0, false, false);
        acc1 = __builtin_amdgcn_wmma_f32_16x16x4_f32(false, a, false, b1,
                                                     (short)0, acc1, false, false);
        acc2 = __builtin_amdgcn_wmma_f32_16x16x4_f32(false, a, false, b2,
                                                     (short)0, acc2, false, false);
        acc3 = __builtin_amdgcn_wmma_f32_16x16x4_f32(false, a, false, b3,
                                                     (short)0, acc3, false, false);
    }

    int mstore = m0 + ((lane >> 4) << 3);   // +8 rows for upper half-wave
    float bb0 = bl[ 0 + l], bb1 = bl[16 + l], bb2 = bl[32 + l], bb3 = bl[48 + l];
#pragma unroll
    for (int r = 0; r < 8; ++r) {
        float* orow = out + (long long)(mstore + r) * FEAT;
        orow[ 0 + l] = acc0[r] + bb0;
        orow[16 + l] = acc1[r] + bb1;
        orow[32 + l] = acc2[r] + bb2;
        orow[48 + l] = acc3[r] + bb3;
    }
}

// Scalar tail for n % 16 rows (never launched for N=65536; kept for safety).
__global__ void sgc_gemm_tail_kernel(const float* __restrict__ h,
                                     const float* __restrict__ W,
                                     const float* __restrict__ bias,
                                     float* __restrict__ out, int start, int n) {
    int i = start + blockIdx.x;
    int j = threadIdx.x;  // 64 threads
    if (i >= n) return;
    float s = 0.0f;
    for (int k = 0; k < FEAT; ++k) s += h[(long long)i * FEAT + k] * W[j * FEAT + k];
    __syncthreads();  // all reads of row i done before in-place writes
    out[(long long)i * FEAT + j] = s + bias[j];
}

// ---------------------------------------------------------------------------
extern "C" void kernel_launch(void* const* d_in, const int* in_sizes, int n_in,
                              void* d_out, int out_size, void* d_ws, size_t ws_size,
                              hipStream_t stream) {
    const float*      x  = (const float*)d_in[0];
    const long long*  ei = (const long long*)d_in[1];   // int64 edge_index [2, E]
    const float*      W  = (const float*)d_in[2];
    const float*      b  = (const float*)d_in[3];

    int n = in_sizes[0] / FEAT;
    int e = in_sizes[1] / 2;

    const long long* row = ei;
    const long long* col = ei + e;

    float* dinv = (float*)d_ws;          // n floats (deg, then deg^-1/2)
    float* htmp = dinv + n;              // n*FEAT floats
    float* out  = (float*)d_out;         // also serves as h after hop 2

    int tb = 256;
    // 1) degrees (over col, with self loop) -> deg^{-1/2}
    deg_init_kernel<<<(n + tb - 1) / tb, tb, 0, stream>>>(dinv, n);
    deg_count_kernel<<<(e + tb - 1) / tb, tb, 0, stream>>>(col, dinv, e);
    dinv_kernel<<<(n + tb - 1) / tb, tb, 0, stream>>>(dinv, n);

    int elem_blocks = (n * FEAT + tb - 1) / tb;
    int edge_blocks = (e + (tb / FEAT) - 1) / (tb / FEAT);

    // 2) hop 1: x -> htmp
    prop_init_kernel<<<elem_blocks, tb, 0, stream>>>(x, dinv, htmp, n);
    prop_edge_kernel<<<edge_blocks, tb, 0, stream>>>(row, col, dinv, x, htmp, e);

    // 3) hop 2: htmp -> out (d_out doubles as h buffer)
    prop_init_kernel<<<elem_blocks, tb, 0, stream>>>(htmp, dinv, out, n);
    prop_edge_kernel<<<edge_blocks, tb, 0, stream>>>(row, col, dinv, htmp, out, e);

    // 4) out = out @ W^T + b, in place, fp32 WMMA
    int gemm_blocks = (n + 127) / 128;
    sgc_gemm_wmma_kernel<<<gemm_blocks, 256, 0, stream>>>(out, W, b, out, n);

    int covered = (n / 16) * 16;
    if (covered < n) {
        sgc_gemm_tail_kernel<<<n - covered, FEAT, 0, stream>>>(out, W, b, out,
                                                               covered, n);
    }
}